// OccGridEstimator_1726576856128
// MI455X (gfx1250) — compile-verified
//
#include <hip/hip_runtime.h>
#include <hip/hip_bf16.h>
#include <math.h>

typedef float v2f __attribute__((ext_vector_type(2)));
typedef float v8f __attribute__((ext_vector_type(8)));

#define N_RAYS    32768
#define N_SAMPLES 256
#define RESG      128

// One wave32 per ray. 256 samples per ray form a 16x16 f32 matrix X
// (row r = samples 16r..16r+15). Exclusive flat scan:
//   S = Lstrict * (X * J) + X * Ustrict
// computed with chained V_WMMA_F32_16X16X4_F32 (K=4 per issue, 4 per product).
__global__ __launch_bounds__(256, 1)
void occ_march_wmma(const float* __restrict__ rays_o,
                    const float* __restrict__ rays_d,
                    const float* __restrict__ grid,
                    float* __restrict__ out)
{
    constexpr float STEP = 1.7320508075688772f / 256.0f; // sqrt(3)/256
    const int lane = (int)(threadIdx.x & 31u);
    const int wid  = (int)(threadIdx.x >> 5u);
    const int ray  = (int)blockIdx.x * 8 + wid;

    // ---- per-ray setup (uniform across the wave) ----
    float ox = rays_o[3 * ray + 0], oy = rays_o[3 * ray + 1], oz = rays_o[3 * ray + 2];
    float dx = rays_d[3 * ray + 0], dy = rays_d[3 * ray + 1], dz = rays_d[3 * ray + 2];
    float inv = 1.0f / sqrtf(dx * dx + dy * dy + dz * dz);
    dx *= inv; dy *= inv; dz *= inv;
    float sdx = (fabsf(dx) < 1e-8f) ? 1e-8f : dx;
    float sdy = (fabsf(dy) < 1e-8f) ? 1e-8f : dy;
    float sdz = (fabsf(dz) < 1e-8f) ? 1e-8f : dz;
    float t0x = (0.0f - ox) / sdx, t1x = (1.0f - ox) / sdx;
    float t0y = (0.0f - oy) / sdy, t1y = (1.0f - oy) / sdy;
    float t0z = (0.0f - oz) / sdz, t1z = (1.0f - oz) / sdz;
    float tmin = fmaxf(fmaxf(fminf(t0x, t1x), fminf(t0y, t1y)), fminf(t0z, t1z));
    tmin = fmaxf(tmin, 0.0f);
    float tmax = fminf(fminf(fmaxf(t0x, t1x), fmaxf(t0y, t1y)), fmaxf(t0z, t1z));

    // w(s) = sigma*STEP if (inside && occupied) else 0;  1-alpha = exp(-w) exactly.
    auto sample_w = [&](int s) -> float {
        float tmid = tmin + ((float)s + 0.5f) * STEP;
        float px = ox + tmid * dx;
        float py = oy + tmid * dy;
        float pz = oz + tmid * dz;
        int ix = (int)floorf(px * (float)RESG);
        int iy = (int)floorf(py * (float)RESG);
        int iz = (int)floorf(pz * (float)RESG);
        ix = ix < 0 ? 0 : (ix > RESG - 1 ? RESG - 1 : ix);
        iy = iy < 0 ? 0 : (iy > RESG - 1 ? RESG - 1 : iy);
        iz = iz < 0 ? 0 : (iz > RESG - 1 ? RESG - 1 : iz);
        float sigma = 10.0f * grid[((size_t)ix * RESG + iy) * RESG + iz];
        bool inside = (tmid < tmax) && (tmax > tmin);
        bool valid  = inside && (sigma > 5.0f);   // OCC_THRE = 0.5*10
        return valid ? sigma * STEP : 0.0f;
    };

    const int m    = lane & 15;               // A row / D column index
    const int n    = lane & 15;
    const int base = (lane < 16) ? 0 : 2;     // K sub-pair owned by this half-wave
    const bool lo  = (lane < 16);

    // ---- build X as four 16x4 A-operand chunks: xa[k] = X[:, 4k+base .. 4k+base+1] ----
    v2f xa[4];
    #pragma unroll
    for (int k = 0; k < 4; ++k) {
        xa[k].x = sample_w(16 * m + 4 * k + base + 0);
        xa[k].y = sample_w(16 * m + 4 * k + base + 1);
    }

    v2f bones; bones.x = 1.0f; bones.y = 1.0f;      // rows of J (all ones)
    v8f accY = {};                                  // Y = X * J   (row sums broadcast)
    v8f accS = {};                                  // S1 = X * Ustrict, then += Lstrict*Y

    #pragma unroll
    for (int k = 0; k < 4; ++k) {
        accY = __builtin_amdgcn_wmma_f32_16x16x4_f32(false, xa[k], false, bones,
                                                     (short)0, accY, false, false);
        v2f bu; // Ustrict[4k+base+j, n] = (row < n)
        bu.x = ((4 * k + base + 0) < n) ? 1.0f : 0.0f;
        bu.y = ((4 * k + base + 1) < n) ? 1.0f : 0.0f;
        accS = __builtin_amdgcn_wmma_f32_16x16x4_f32(false, xa[k], false, bu,
                                                     (short)0, accS, false, false);
    }

    // ---- re-layout Y (D layout) into B-operand chunks via half-wave swap ----
    float yv[8], swv[8];
    #pragma unroll
    for (int v = 0; v < 8; ++v) {
        yv[v]  = accY[v];
        swv[v] = __shfl_xor(yv[v], 16, 32);
    }
    v2f by[4];
    by[0].x = lo ? yv[0] : swv[2];  by[0].y = lo ? yv[1] : swv[3];
    by[1].x = lo ? yv[4] : swv[6];  by[1].y = lo ? yv[5] : swv[7];
    by[2].x = lo ? swv[0] : yv[2];  by[2].y = lo ? swv[1] : yv[3];
    by[3].x = lo ? swv[4] : yv[6];  by[3].y = lo ? swv[5] : yv[7];

    #pragma unroll
    for (int k = 0; k < 4; ++k) {
        v2f la; // Lstrict[m, 4k+base+j] = (m > col)
        la.x = (m > (4 * k + base + 0)) ? 1.0f : 0.0f;
        la.y = (m > (4 * k + base + 1)) ? 1.0f : 0.0f;
        accS = __builtin_amdgcn_wmma_f32_16x16x4_f32(false, la, false, by[k],
                                                     (short)0, accS, false, false);
    }

    // ---- emit outputs from the D layout: lane holds S[v+8*hi, n] ----
    const int hi = lane >> 4;
    const size_t PLANE = (size_t)N_RAYS * N_SAMPLES;
    float* o_ts = out;
    float* o_te = out + PLANE;
    float* o_tr = out + 2 * PLANE;
    float* o_vi = out + 3 * PLANE;

    #pragma unroll
    for (int v = 0; v < 8; ++v) {
        int s = 16 * (v + 8 * hi) + n;
        float wv    = sample_w(s);              // recompute at D position (L2-resident)
        float S     = accS[v];                  // exclusive cumsum of w before s
        float trans = __expf(-S);
        float alpha = 1.0f - __expf(-wv);
        float ts    = tmin + (float)s * STEP;
        float vis   = ((wv > 0.0f) && (trans > 1e-4f) && (alpha > 1e-4f)) ? 1.0f : 0.0f;
        size_t g = (size_t)ray * N_SAMPLES + s;
        o_ts[g] = ts;
        o_te[g] = ts + STEP;
        o_tr[g] = trans;
        o_vi[g] = vis;
    }
}

extern "C" void kernel_launch(void* const* d_in, const int* in_sizes, int n_in,
                              void* d_out, int out_size, void* d_ws, size_t ws_size,
                              hipStream_t stream) {
    (void)in_sizes; (void)n_in; (void)out_size; (void)d_ws; (void)ws_size;
    const float* rays_o = (const float*)d_in[0];   // [32768, 3]
    const float* rays_d = (const float*)d_in[1];   // [32768, 3]
    const float* grid   = (const float*)d_in[2];   // [128, 128, 128]
    float* out = (float*)d_out;                    // 4 planes of 32768*256 f32
    occ_march_wmma<<<dim3(N_RAYS / 8), dim3(256), 0, stream>>>(rays_o, rays_d, grid, out);
}